// signalEncoding_72885595013488
// MI455X (gfx1250) — compile-verified
//
#include <hip/hip_runtime.h>
#include <math.h>

// ---------------------------------------------------------------------------
// Fused signal-encoding pipeline for MI455X (gfx1250, wave32).
// Bandwidth-bound: ~246 MB HBM traffic -> ~10.6 us floor @ 23.3 TB/s.
// Each depthwise circular conv pair composes into ONE 25x25 circulant per
// signal, executed on the matrix pipe via V_WMMA_F32_16X16X4_F32
// (full fp32, 16 batch rows per wave, 7 k-steps, two overlapping N-tiles).
// ---------------------------------------------------------------------------

typedef float v2f __attribute__((ext_vector_type(2)));
typedef float v8f __attribute__((ext_vector_type(8)));

#define S_NUM 75
#define L_NUM 25
#define BTILE 16          // batch rows per wave / WMMA M dim
#define STILE 8           // signals (waves) per workgroup
#define KPAD  28          // K dim padded to 7 wmma k-steps of 4
#define TILE1_OFF 9       // second N-tile covers l = 9..24 (overlap, no guards)
#define WPK_PER_SIG 448   // 7 ksteps * 2 lane-halves * 32 cols (float2 entries)
#define PER   (STILE * L_NUM)   // 200
#define TOT   (BTILE * PER)     // 3200

// ---------------- composed circulant weight pre-pass -----------------------
// conv pair a,b (k=11, dil=1, circular pad 5):
//   y[l] = sum_{i,j} wa[i] wb[j] x[(l+i+j-10) mod 25] + (b_a*sum(wb) + b_b)
// => M1[l][k] = sum_i wa[i] * wb[(k-l+10-i) mod 25]  (tap valid if in [0,10])
// conv pair (k=9, dil=2, pad 8): offsets 2(i+j)-16; 2^{-1} mod 25 = 13.
__device__ __forceinline__ float mval1(int l, int k, const float* wa, const float* wb) {
  if (k >= L_NUM) return 0.f;
  float acc = 0.f;
#pragma unroll
  for (int i = 0; i < 11; ++i) {
    int d = (k - l + 10 - i + 50) % 25;        // always non-negative
    if (d <= 10) acc += wa[i] * wb[d];
  }
  return acc;
}
__device__ __forceinline__ float mval2(int l, int k, const float* wa, const float* wb) {
  if (k >= L_NUM) return 0.f;
  float acc = 0.f;
#pragma unroll
  for (int i = 0; i < 9; ++i) {
    int r = (k - l + 16 - 2 * i + 50) % 25;
    int j = (13 * r) % 25;                     // solve 2j == r (mod 25)
    if (j <= 8) acc += wa[i] * wb[j];
  }
  return acc;
}

// W layout (per signal): float2 entry e, e = (t*2+half)*32 + c
//   c in [0,16):  l = c        (N-tile 0, l = 0..15)
//   c in [16,32): l = c - 7    (N-tile 1, l = 9..24)
//   .x = M[l][4t+2*half], .y = M[l][4t+2*half+1]
// i.e. exactly one lane's B-operand register pair for k-step t.
__global__ void build_weights(const float* __restrict__ w1a, const float* __restrict__ b1a,
                              const float* __restrict__ w1b, const float* __restrict__ b1b,
                              const float* __restrict__ w2a, const float* __restrict__ b2a,
                              const float* __restrict__ w2b, const float* __restrict__ b2b,
                              float2* __restrict__ W1, float2* __restrict__ W2,
                              float* __restrict__ c1, float* __restrict__ c2) {
  const int s = blockIdx.x;
  const float* a1 = w1a + s * 11;
  const float* v1 = w1b + s * 11;
  const float* a2 = w2a + s * 9;
  const float* v2 = w2b + s * 9;
  for (int e = threadIdx.x; e < WPK_PER_SIG; e += blockDim.x) {
    const int c = e & 31;
    const int l = (c < 16) ? c : (c - (16 - TILE1_OFF));
    const int th = e >> 5;        // t*2 + half
    const int k0 = th * 2;        // 4t + 2*half
    float2 m1, m2;
    m1.x = mval1(l, k0, a1, v1);
    m1.y = mval1(l, k0 + 1, a1, v1);
    m2.x = mval2(l, k0, a2, v2);
    m2.y = mval2(l, k0 + 1, a2, v2);
    W1[(size_t)s * WPK_PER_SIG + e] = m1;
    W2[(size_t)s * WPK_PER_SIG + e] = m2;
  }
  if (threadIdx.x == 0) {
    float sw1 = 0.f, sw2 = 0.f;
#pragma unroll
    for (int j = 0; j < 11; ++j) sw1 += v1[j];
#pragma unroll
    for (int j = 0; j < 9; ++j) sw2 += v2[j];
    c1[s] = b1a[s] * sw1 + b1b[s];
    c2[s] = b2a[s] * sw2 + b2b[s];
  }
}

// ---------------- fused main kernel ----------------------------------------
// One conv-pair as a 16x25 @ 25x25 product on the matrix pipe.
// A (16x4 f32, ISA layout): lane = half*16+row holds X[row][4t+2half .. +1].
// B (4x16 f32, mirrored layout): lane holds M[l(col)][4t+2half .. +1].
// D (16x16 f32): acc[v] = Y[batch = v+8*half][l(col)].
__device__ __forceinline__ void conv_tile(const float* __restrict__ xb,
                                          float* __restrict__ yb,
                                          const float2* __restrict__ W,
                                          int sigl, int lane) {
  const int half = lane >> 4;
  const int rc = lane & 15;                     // A row / B,D column
  const float* xrow = xb + (sigl * BTILE + rc) * KPAD;
  v8f acc0 = {0.f, 0.f, 0.f, 0.f, 0.f, 0.f, 0.f, 0.f};
  v8f acc1 = {0.f, 0.f, 0.f, 0.f, 0.f, 0.f, 0.f, 0.f};
#pragma unroll
  for (int t = 0; t < 7; ++t) {
    const int kk = 4 * t + 2 * half;
    const v2f a = *(const v2f*)(xrow + kk);            // ds_load_b64
    const float2 w0 = W[(t * 2 + half) * 32 + rc];      // global_load_b64, L2-hot
    const float2 w1 = W[(t * 2 + half) * 32 + 16 + rc];
    const v2f bv0 = {w0.x, w0.y};
    const v2f bv1 = {w1.x, w1.y};
    acc0 = __builtin_amdgcn_wmma_f32_16x16x4_f32(false, a, false, bv0, (short)0, acc0, false, false);
    acc1 = __builtin_amdgcn_wmma_f32_16x16x4_f32(false, a, false, bv1, (short)0, acc1, false, false);
  }
  // Unconditional stores: tile0 -> cols 0..15, tile1 -> cols 9..24.
  // Overlap cols 9..15 receive bitwise-identical values (same k-order).
#pragma unroll
  for (int v = 0; v < 8; ++v) {
    const int brow = v + 8 * half;
    float* yr = yb + (sigl * BTILE + brow) * KPAD;
    yr[rc] = acc0[v];
    yr[TILE1_OFF + rc] = acc1[v];
  }
}

// Per-(batch,signal)-row epilogue: +bias, LayerNorm over L, SELU, residual*0.5.
__device__ __forceinline__ void row_post(const float* __restrict__ yrow,
                                         const float* __restrict__ xrow,
                                         float* __restrict__ drow,
                                         float cadd,
                                         const float* __restrict__ g,
                                         const float* __restrict__ be) {
  float y[L_NUM];
  float mu = 0.f;
#pragma unroll
  for (int l = 0; l < L_NUM; ++l) { y[l] = yrow[l] + cadd; mu += y[l]; }
  mu *= (1.0f / L_NUM);
  float var = 0.f;
#pragma unroll
  for (int l = 0; l < L_NUM; ++l) { const float d = y[l] - mu; var += d * d; }
  var *= (1.0f / L_NUM);
  const float inv = 1.0f / sqrtf(var + 1e-10f);
  const float SC = 1.0507009873554805f, AL = 1.6732632423543772f;
#pragma unroll
  for (int l = 0; l < L_NUM; ++l) {
    const float n = (y[l] - mu) * inv * g[l] + be[l];
    const float se = (n > 0.f) ? SC * n : SC * AL * (__expf(n) - 1.0f);
    drow[l] = 0.5f * (xrow[l] + se);
  }
}

__global__ __launch_bounds__(256) void fused_kernel(
    const float* __restrict__ in,
    const float2* __restrict__ W1, const float2* __restrict__ W2,
    const float* __restrict__ c1, const float* __restrict__ c2,
    const float* __restrict__ g1, const float* __restrict__ be1,
    const float* __restrict__ g2, const float* __restrict__ be2,
    float* __restrict__ out) {
  __shared__ float xb[STILE * BTILE * KPAD];   // X, then h (K-padded rows)
  __shared__ float yb[STILE * BTILE * KPAD];   // conv results, then final out
  __shared__ float pb[4 * PER + 2 * STILE];    // g1|be1|g2|be2 tiles + c1|c2

  const int tid = threadIdx.x;
  const int lane = tid & 31;
  const int wid = tid >> 5;
  const int b0 = blockIdx.x * BTILE;
  const int s0 = blockIdx.y * STILE;
  const int nsig = (S_NUM - s0 < STILE) ? (S_NUM - s0) : STILE;

  // zero only the K-padding strip (cols 24..27 of every row; col 24 is
  // overwritten by real data for active rows) -- shift-only index math
  for (int i = tid; i < STILE * BTILE * 4; i += 256)
    xb[(i >> 2) * KPAD + 24 + (i & 3)] = 0.f;

  // stage LN params + folded biases to LDS (shared across the 16 batch rows)
  for (int i = tid; i < 4 * PER; i += 256) {
    const int a = i / PER;                 // constant divisions -> mulhi
    const int rem = i - a * PER;
    const int sg = rem / L_NUM;
    const int l = rem - sg * L_NUM;
    if (sg < nsig) {
      const float* src = (a == 0) ? g1 : (a == 1) ? be1 : (a == 2) ? g2 : be2;
      pb[i] = src[(s0 + sg) * L_NUM + l];
    }
  }
  if (tid < 2 * STILE) {
    const int which = tid >> 3, sg = tid & (STILE - 1);
    if (sg < nsig) pb[4 * PER + tid] = (which ? c2 : c1)[s0 + sg];
  }

  // coalesced load: signals are contiguous -> 200-float runs per batch row
  for (int i = tid; i < TOT; i += 256) {
    const int b = i / PER;                 // constant divisions
    const int rem = i - b * PER;
    const int sg = rem / L_NUM;
    const int l = rem - sg * L_NUM;
    if (sg < nsig)
      xb[(sg * BTILE + b) * KPAD + l] =
          in[((size_t)(b0 + b) * S_NUM + (s0 + sg)) * L_NUM + l];
  }
  __syncthreads();

  // conv pair 1 (composed circulant) on the matrix pipe
  if (wid < nsig)
    conv_tile(xb, yb, W1 + (size_t)(s0 + wid) * WPK_PER_SIG, wid, lane);
  __syncthreads();

  // LN1 + SELU + h = (x + y1)/2  -> overwrite X buffer (padding cols stay 0)
  if (tid < STILE * BTILE) {
    const int sg = tid & (STILE - 1);
    const int b = tid >> 3;
    if (sg < nsig) {
      float* xr = xb + (sg * BTILE + b) * KPAD;
      row_post(yb + (sg * BTILE + b) * KPAD, xr, xr, pb[4 * PER + sg],
               pb + 0 * PER + sg * L_NUM, pb + 1 * PER + sg * L_NUM);
    }
  }
  __syncthreads();

  // conv pair 2 on h
  if (wid < nsig)
    conv_tile(xb, yb, W2 + (size_t)(s0 + wid) * WPK_PER_SIG, wid, lane);
  __syncthreads();

  // LN2 + SELU + out = (h + y2)/2 -> into yb
  if (tid < STILE * BTILE) {
    const int sg = tid & (STILE - 1);
    const int b = tid >> 3;
    if (sg < nsig) {
      float* yr = yb + (sg * BTILE + b) * KPAD;
      row_post(yr, xb + (sg * BTILE + b) * KPAD, yr, pb[4 * PER + STILE + sg],
               pb + 2 * PER + sg * L_NUM, pb + 3 * PER + sg * L_NUM);
    }
  }
  __syncthreads();

  // coalesced store
  for (int i = tid; i < TOT; i += 256) {
    const int b = i / PER;
    const int rem = i - b * PER;
    const int sg = rem / L_NUM;
    const int l = rem - sg * L_NUM;
    if (sg < nsig)
      out[((size_t)(b0 + b) * S_NUM + (s0 + sg)) * L_NUM + l] =
          yb[(sg * BTILE + b) * KPAD + l];
  }
}

// ---------------------------------------------------------------------------
extern "C" void kernel_launch(void* const* d_in, const int* in_sizes, int n_in,
                              void* d_out, int out_size, void* d_ws, size_t ws_size,
                              hipStream_t stream) {
  const float* in  = (const float*)d_in[0];
  const float* w1a = (const float*)d_in[1];
  const float* b1a = (const float*)d_in[2];
  const float* w1b = (const float*)d_in[3];
  const float* b1b = (const float*)d_in[4];
  const float* g1  = (const float*)d_in[5];
  const float* be1 = (const float*)d_in[6];
  const float* w2a = (const float*)d_in[7];
  const float* b2a = (const float*)d_in[8];
  const float* w2b = (const float*)d_in[9];
  const float* b2b = (const float*)d_in[10];
  const float* g2  = (const float*)d_in[11];
  const float* be2 = (const float*)d_in[12];

  const int Btot = in_sizes[0] / (S_NUM * L_NUM);   // 16384

  // workspace: pre-swizzled composed circulants + folded biases (~538 KB)
  float2* W1 = (float2*)d_ws;
  float2* W2 = W1 + (size_t)S_NUM * WPK_PER_SIG;
  float* c1 = (float*)(W2 + (size_t)S_NUM * WPK_PER_SIG);
  float* c2 = c1 + S_NUM;

  build_weights<<<S_NUM, 256, 0, stream>>>(w1a, b1a, w1b, b1b, w2a, b2a, w2b, b2b,
                                           W1, W2, c1, c2);

  dim3 grid(Btot / BTILE, (S_NUM + STILE - 1) / STILE);
  fused_kernel<<<grid, 256, 0, stream>>>(in, W1, W2, c1, c2, g1, be1, g2, be2,
                                         (float*)d_out);
}